// Decoder_1649267441942
// MI455X (gfx1250) — compile-verified
//
#include <hip/hip_runtime.h>

#define Bb 4096
#define Tt 168
#define Dd 64
#define Hh 128
#define Gg 384  // 3*H

typedef __attribute__((ext_vector_type(2))) float v2f;
typedef __attribute__((ext_vector_type(8))) float v8f;

__device__ __forceinline__ v8f wmma4(v2f a, v2f b, v8f c) {
  // D = A(16x4) * B(4x16) + C(16x16), full f32
  return __builtin_amdgcn_wmma_f32_16x16x4_f32(false, a, false, b, (short)0, c,
                                               false, false);
}

__global__ __launch_bounds__(256, 1) void gru_decoder_fused(
    const float* __restrict__ ff,   // (B,T,D)
    const float* __restrict__ h0,   // (1,B,H)
    const float* __restrict__ y0,   // (B,)
    const float* __restrict__ W_ih, // (3H, D+1)
    const float* __restrict__ W_hh, // (3H, H)
    const float* __restrict__ b_ih, // (3H,)
    const float* __restrict__ b_hh, // (3H,)
    const float* __restrict__ Wo,   // (1,H)
    const float* __restrict__ bo,   // (1,)
    float* __restrict__ out)        // (B,T)
{
  extern __shared__ float smem[];
  float* whh   = smem;               // 384*128 = 49152
  float* wih   = whh + Gg * Hh;      // 384*64  = 24576
  float* hbuf  = wih + Gg * Dd;      // 32*128  = 4096
  float* wy    = hbuf + 32 * Hh;     // 384
  float* bih   = wy + Gg;            // 384
  float* bhh   = bih + Gg;           // 384
  float* woL   = bhh + Gg;           // 128
  float* yprev = woL + Hh;           // 32
  float* yacc  = yprev + 32;         // 32     -> total 79168 floats = 316672 B

  const int tid  = threadIdx.x;
  const int lane = tid & 31;
  const int wave = tid >> 5;
  const int msub = wave >> 2;        // which 16-row M-subtile (0/1)
  const int cgB  = (wave & 3) * 2;   // first of this wave's 2 column groups
  const int lo16 = lane & 15;
  const int hi   = lane >> 4;        // K-pair half of the wave
  const int kb   = hi * 2;
  const int b0   = blockIdx.x * 32;

  // ---- stage weights + initial state into LDS (once; reused for 168 steps)
  for (int i = tid; i < Gg * Hh; i += 256) whh[i] = W_hh[i];
  for (int i = tid; i < Gg * Dd; i += 256)
    wih[i] = W_ih[(i >> 6) * (Dd + 1) + (i & 63)];
  for (int i = tid; i < 32 * Hh; i += 256)
    hbuf[i] = h0[(size_t)(b0 + (i >> 7)) * Hh + (i & 127)];
  for (int i = tid; i < Gg; i += 256) {
    wy[i]  = W_ih[i * (Dd + 1) + Dd];  // y-feedback column of W_ih
    bih[i] = b_ih[i];
    bhh[i] = b_hh[i];
  }
  if (tid < Hh) woL[tid] = Wo[tid];
  if (tid < 32) { yprev[tid] = y0[b0 + tid]; yacc[tid] = 0.0f; }
  const float bo0 = bo[0];
  __syncthreads();

  const int rowA = msub * 16 + lo16;                  // A-fragment M row
  const float* ffbase = ff + (size_t)(b0 + rowA) * Tt * Dd;
  const float* hrow   = hbuf + rowA * Hh;

  for (int t = 0; t < Tt; ++t) {
    const float* ffr = ffbase + (size_t)t * Dd;

    // Issue all 16 A-fragment loads of this step's gx GEMM back-to-back, then
    // prefetch next step's row (global_prefetch_b8) so HBM latency overlaps
    // this step's 288 WMMAs per wave.
    v2f af[16];
#pragma unroll
    for (int k0 = 0; k0 < Dd; k0 += 4) {
      af[k0 >> 2].x = ffr[k0 + kb];
      af[k0 >> 2].y = ffr[k0 + kb + 1];
    }
    if (t + 1 < Tt) __builtin_prefetch(ffr + Dd, 0, 0);

    float ypv[8];
#pragma unroll
    for (int v = 0; v < 8; ++v) ypv[v] = yprev[msub * 16 + v + 8 * hi];

    // accumulators: r-gate, z-gate, gx-part of n, gh-part of n (kept separate
    // because n = tanh(gx_n + r * gh_n))
    v8f accR[2], accZ[2], accGXN[2], accGHN[2];
#pragma unroll
    for (int cg = 0; cg < 2; ++cg) {
      const int hc = (cgB + cg) * 16 + lo16;
      const float biR = bih[hc] + bhh[hc];
      const float biZ = bih[Hh + hc] + bhh[Hh + hc];
      const float biN = bih[2 * Hh + hc];
      const float bhN = bhh[2 * Hh + hc];
      const float wyR = wy[hc], wyZ = wy[Hh + hc], wyN = wy[2 * Hh + hc];
#pragma unroll
      for (int v = 0; v < 8; ++v) {
        accR[cg][v]   = biR + ypv[v] * wyR;
        accZ[cg][v]   = biZ + ypv[v] * wyZ;
        accGXN[cg][v] = biN + ypv[v] * wyN;
        accGHN[cg][v] = bhN;
      }
    }

    // ---- gx GEMM: (16 x 64) @ (64 x 96-per-wave), A in regs, B from LDS.
    // FULLY unrolled so af[] indices are constants (no m0/v_movrels).
#pragma unroll
    for (int k0 = 0; k0 < Dd; k0 += 4) {
      const v2f a = af[k0 >> 2];
#pragma unroll
      for (int cg = 0; cg < 2; ++cg) {
        const int gc = (cgB + cg) * 16 + lo16;
        v2f bR, bZ, bN;
        bR.x = wih[gc * Dd + k0 + kb];            bR.y = wih[gc * Dd + k0 + kb + 1];
        bZ.x = wih[(Hh + gc) * Dd + k0 + kb];     bZ.y = wih[(Hh + gc) * Dd + k0 + kb + 1];
        bN.x = wih[(2 * Hh + gc) * Dd + k0 + kb]; bN.y = wih[(2 * Hh + gc) * Dd + k0 + kb + 1];
        accR[cg]   = wmma4(a, bR, accR[cg]);
        accZ[cg]   = wmma4(a, bZ, accZ[cg]);
        accGXN[cg] = wmma4(a, bN, accGXN[cg]);
      }
    }

    // ---- gh GEMM: (16 x 128) @ (128 x 96-per-wave), all operands in LDS
#pragma unroll 4
    for (int k0 = 0; k0 < Hh; k0 += 4) {
      v2f a; a.x = hrow[k0 + kb]; a.y = hrow[k0 + kb + 1];
#pragma unroll
      for (int cg = 0; cg < 2; ++cg) {
        const int gc = (cgB + cg) * 16 + lo16;
        v2f bR, bZ, bN;
        bR.x = whh[gc * Hh + k0 + kb];            bR.y = whh[gc * Hh + k0 + kb + 1];
        bZ.x = whh[(Hh + gc) * Hh + k0 + kb];     bZ.y = whh[(Hh + gc) * Hh + k0 + kb + 1];
        bN.x = whh[(2 * Hh + gc) * Hh + k0 + kb]; bN.y = whh[(2 * Hh + gc) * Hh + k0 + kb + 1];
        accR[cg]   = wmma4(a, bR, accR[cg]);
        accZ[cg]   = wmma4(a, bZ, accZ[cg]);
        accGHN[cg] = wmma4(a, bN, accGHN[cg]);
      }
    }

    // ---- snapshot h_old (C/D layout positions) before it is overwritten
    float hold[2][8];
#pragma unroll
    for (int cg = 0; cg < 2; ++cg)
#pragma unroll
      for (int v = 0; v < 8; ++v)
        hold[cg][v] = hbuf[(msub * 16 + v + 8 * hi) * Hh + (cgB + cg) * 16 + lo16];

    __syncthreads();  // all reads of hbuf/yprev complete

    // ---- gates + state update + readout partials (all wave-local)
    float p[8];
#pragma unroll
    for (int v = 0; v < 8; ++v) p[v] = 0.0f;

#pragma unroll
    for (int cg = 0; cg < 2; ++cg) {
      const int hc = (cgB + cg) * 16 + lo16;
      const float wov = woL[hc];
#pragma unroll
      for (int v = 0; v < 8; ++v) {
        const float r  = 1.0f / (1.0f + __expf(-accR[cg][v]));
        const float z  = 1.0f / (1.0f + __expf(-accZ[cg][v]));
        const float nn = tanhf(accGXN[cg][v] + r * accGHN[cg][v]);
        const float hn = (1.0f - z) * nn + z * hold[cg][v];
        hbuf[(msub * 16 + v + 8 * hi) * Hh + hc] = hn;
        p[v] += wov * hn;
      }
    }

    // reduce readout partials across the 16 lanes of each half-wave
#pragma unroll
    for (int mask = 1; mask < 16; mask <<= 1)
#pragma unroll
      for (int v = 0; v < 8; ++v) p[v] += __shfl_xor(p[v], mask, 32);

    if (lo16 == 0) {
#pragma unroll
      for (int v = 0; v < 8; ++v)
        atomicAdd(&yacc[msub * 16 + 8 * hi + v], p[v]);  // ds_add_f32
    }
    __syncthreads();  // h_new stored + yacc complete

    if (tid < 32) {
      const float y = yacc[tid] + bo0;
      out[(size_t)(b0 + tid) * Tt + t] = y;
      yprev[tid] = y;
      yacc[tid]  = 0.0f;
    }
    __syncthreads();  // yprev/yacc ready for next step
  }
}

extern "C" void kernel_launch(void* const* d_in, const int* in_sizes, int n_in,
                              void* d_out, int out_size, void* d_ws,
                              size_t ws_size, hipStream_t stream) {
  const float* ff   = (const float*)d_in[0];
  const float* h0   = (const float*)d_in[1];
  const float* y0   = (const float*)d_in[2];
  const float* W_ih = (const float*)d_in[3];
  const float* W_hh = (const float*)d_in[4];
  const float* b_ih = (const float*)d_in[5];
  const float* b_hh = (const float*)d_in[6];
  const float* Wo   = (const float*)d_in[7];
  const float* bo   = (const float*)d_in[8];
  float* out = (float*)d_out;

  const size_t lds_bytes =
      (size_t)(Gg * Hh + Gg * Dd + 32 * Hh + 3 * Gg + Hh + 64) * sizeof(float);

  gru_decoder_fused<<<Bb / 32, 256, lds_bytes, stream>>>(
      ff, h0, y0, W_ih, W_hh, b_ih, b_hh, Wo, bo, out);
}